// NGCF_18365280158072
// MI455X (gfx1250) — compile-verified
//
#include <hip/hip_runtime.h>
#include <hip/hip_bf16.h>

// ---------------------------------------------------------------------------
// NGCF forward for MI455X (gfx1250, wave32, WMMA).
// Memory-bound workload (~600 MB traffic -> ~26 us @ 23.3 TB/s); dense GEMMs
// (7.4 GFLOP total) are free, so we keep full fp32 accuracy and use
// V_WMMA_F32_16X16X4_F32 for the [*,64]x[64,64] projections.
// Weights are staged TRANSPOSED in LDS (padded stride 66) so each WMMA
// B-fragment is one contiguous ds_load_b64 into an even VGPR pair.
// ---------------------------------------------------------------------------

#define N_USER 50000
#define N_ITEM 100000
#define NTOT   (N_USER + N_ITEM)   // 150000 nodes, divisible by 16
#define NEDGE  2400000
#define DD     64
#define SLOPE  0.2f
#define NTILES (NTOT / 16)         // 9375
#define WSTR   66                  // padded LDS stride (even -> 8B align; odd/64 -> no bank conflicts)

typedef __attribute__((ext_vector_type(2))) float v2f;
typedef __attribute__((ext_vector_type(8))) float v8f;

// ---------------------------------------------------------------------------
// 1) Build ego = [user_emb; item_emb], and write out[:, 0:64] = ego (raw).
// ---------------------------------------------------------------------------
__global__ __launch_bounds__(256) void ngcf_init(const float4* __restrict__ ue,
                                                 const float4* __restrict__ ie,
                                                 float4* __restrict__ ego4,
                                                 float4* __restrict__ out4) {
    size_t i = (size_t)blockIdx.x * blockDim.x + threadIdx.x;
    if (i >= (size_t)NTOT * 16) return;
    size_t row = i >> 4;
    int    c4  = (int)(i & 15);
    float4 v = (row < N_USER) ? ue[row * 16 + c4] : ie[(row - N_USER) * 16 + c4];
    ego4[i] = v;
    out4[row * 64 + c4] = v;   // out row = 256 floats = 64 float4; cols 0..63
}

// ---------------------------------------------------------------------------
// 2) Zero the side accumulator.
// ---------------------------------------------------------------------------
__global__ __launch_bounds__(256) void ngcf_zero(float4* __restrict__ p, size_t n4) {
    size_t i = (size_t)blockIdx.x * blockDim.x + threadIdx.x;
    if (i < n4) p[i] = make_float4(0.f, 0.f, 0.f, 0.f);
}

// ---------------------------------------------------------------------------
// 3) SpMM scatter: side[row] += val * ego[col].  One wave (32 lanes) per edge,
//    each lane owns a float2 of the 64-wide row.  Edge metadata loads are
//    wave-uniform; gather of ego[col] is a coalesced 256B read that stays
//    L2-resident (ego = 38 MB << 192 MB L2); scatter is global_atomic_add_f32.
// ---------------------------------------------------------------------------
__global__ __launch_bounds__(256) void ngcf_spmm(const float* __restrict__ vals,
                                                 const int*  __restrict__ rows,
                                                 const int*  __restrict__ cols,
                                                 const float* __restrict__ ego,
                                                 float* __restrict__ side) {
    size_t tid  = (size_t)blockIdx.x * blockDim.x + threadIdx.x;
    size_t e    = tid >> 5;             // one edge per 32-lane wave
    int    c    = (int)(tid & 31) * 2;  // this lane's float2 column
    if (e >= (size_t)NEDGE) return;
    float v   = vals[e];
    int   col = cols[e];
    int   row = rows[e];
    const float2 s = *(const float2*)(ego + (size_t)col * DD + c);
    float* dst = side + (size_t)row * DD + c;
    atomicAdd(dst + 0, v * s.x);
    atomicAdd(dst + 1, v * s.y);
}

// ---------------------------------------------------------------------------
// 4) Fused layer: per 16-node tile (one wave each):
//      sum = leaky(side @ Wg + bg); bi = leaky((ego*side) @ Wb + bb)
//      ego_new = sum + bi  (written in place over ego)
//      out[:, (k+1)*64 : (k+2)*64] = l2norm(ego_new)
//    GEMM via V_WMMA_F32_16X16X4_F32: K=64 -> 16 K-steps, 4 N-tiles, 2 mats.
//
//    Fragment layouts (ISA 7.12.2, f32):
//      A 16x4 : lane l<16 holds A[l][k0],A[l][k0+1]; lanes 16-31 get k0+2.
//      B 4x16 : lane n<16 holds B[k0][n],B[k0+1][n]; lanes 16-31 get k0+2.
//      C 16x16: vgpr j = row j (lanes 0-15) / row j+8 (lanes 16-31), col=lane&15.
//    LDS weights transposed: sWT[col*WSTR + k] so {B[k][col],B[k+1][col]} is
//    one contiguous 8-byte ds_load_b64 (k even, WSTR even => aligned).
// ---------------------------------------------------------------------------
__global__ __launch_bounds__(256) void ngcf_layer(const float* __restrict__ side,
                                                  float* ego,   // read + in-place write
                                                  const float* __restrict__ Wg,
                                                  const float* __restrict__ bg,
                                                  const float* __restrict__ Wb,
                                                  const float* __restrict__ bb,
                                                  float* __restrict__ out,
                                                  int layer) {
    __shared__ float sWgT[DD * WSTR];
    __shared__ float sWbT[DD * WSTR];
    __shared__ float sbg[DD];
    __shared__ float sbb[DD];

    int tid = threadIdx.x;
    for (int i = tid; i < DD * DD; i += 256) {
        int r = i >> 6;          // weight row (K index)
        int c = i & 63;          // weight col (N index)
        sWgT[c * WSTR + r] = Wg[i];
        sWbT[c * WSTR + r] = Wb[i];
    }
    if (tid < DD) { sbg[tid] = bg[tid]; sbb[tid] = bb[tid]; }
    __syncthreads();

    int wave = tid >> 5;
    int lane = tid & 31;
    int tile = blockIdx.x * 8 + wave;            // 8 waves/block, 1 tile/wave
    if (tile >= NTILES) return;                  // wave-uniform: EXEC stays all-1s
    int row0 = tile * 16;

    int half = lane >> 4;                        // 0: lanes 0-15, 1: lanes 16-31
    int l15  = lane & 15;

    const float* sideRow = side + (size_t)(row0 + l15) * DD + half * 2;
    const float* egoRow  = ego  + (size_t)(row0 + l15) * DD + half * 2;

    v8f accS[4] = {v8f{}, v8f{}, v8f{}, v8f{}};
    v8f accB[4] = {v8f{}, v8f{}, v8f{}, v8f{}};

#pragma unroll
    for (int kk = 0; kk < 16; ++kk) {            // K = 64 = 16 x 4
        float2 sa = *(const float2*)(sideRow + kk * 4);
        float2 ea = *(const float2*)(egoRow  + kk * 4);
        v2f aS; aS.x = sa.x;        aS.y = sa.y;          // side fragment
        v2f aB; aB.x = ea.x * sa.x; aB.y = ea.y * sa.y;   // (ego*side) fragment
        int krow = kk * 4 + half * 2;                     // even
#pragma unroll
        for (int t = 0; t < 4; ++t) {            // 4 N-tiles cover 64 output cols
            int col = t * 16 + l15;
            v2f bS = *(const v2f*)(sWgT + col * WSTR + krow);  // one ds_load_b64
            accS[t] = __builtin_amdgcn_wmma_f32_16x16x4_f32(
                false, aS, false, bS, (short)0, accS[t], false, false);
            v2f bB = *(const v2f*)(sWbT + col * WSTR + krow);  // one ds_load_b64
            accB[t] = __builtin_amdgcn_wmma_f32_16x16x4_f32(
                false, aB, false, bB, (short)0, accB[t], false, false);
        }
    }

    // Epilogue: bias + LeakyReLU(0.2) + combine; per-row L2 norm; store.
    float vals[4][8];
    float sumsq[8];
#pragma unroll
    for (int j = 0; j < 8; ++j) sumsq[j] = 0.f;
#pragma unroll
    for (int t = 0; t < 4; ++t) {
        int col = t * 16 + l15;
        float bgc = sbg[col], bbc = sbb[col];
#pragma unroll
        for (int j = 0; j < 8; ++j) {
            float s = accS[t][j] + bgc;
            s = (s > 0.f) ? s : SLOPE * s;
            float b = accB[t][j] + bbc;
            b = (b > 0.f) ? b : SLOPE * b;
            float e = s + b;
            vals[t][j] = e;
            sumsq[j] += e * e;
        }
    }
    // Reduce across the 16-lane half-group: xor masks 1,2,4,8 never cross the
    // lane-16 boundary, so lanes 0-15 end with row j's norm, 16-31 with row j+8's.
#pragma unroll
    for (int j = 0; j < 8; ++j) {
        float v = sumsq[j];
        v += __shfl_xor(v, 1, 32);
        v += __shfl_xor(v, 2, 32);
        v += __shfl_xor(v, 4, 32);
        v += __shfl_xor(v, 8, 32);
        sumsq[j] = v;
    }

    float* outBase = out + (size_t)row0 * 256 + (size_t)(layer + 1) * 64;
#pragma unroll
    for (int j = 0; j < 8; ++j) {
        int m = j + half * 8;
        float scale = 1.0f / fmaxf(sqrtf(sumsq[j]), 1e-12f);
#pragma unroll
        for (int t = 0; t < 4; ++t) {
            int col = t * 16 + l15;
            float e = vals[t][j];
            ego[(size_t)(row0 + m) * DD + col] = e;          // unnormalized, next layer
            outBase[(size_t)m * 256 + col]     = e * scale;  // normalized slice
        }
    }
}

// ---------------------------------------------------------------------------
// Launch: init -> 3x { zero side; spmm; fused layer }.
// Scratch: ego (38.4 MB) + side (38.4 MB) in d_ws.
// ---------------------------------------------------------------------------
extern "C" void kernel_launch(void* const* d_in, const int* in_sizes, int n_in,
                              void* d_out, int out_size, void* d_ws, size_t ws_size,
                              hipStream_t stream) {
    const float* user_emb = (const float*)d_in[0];
    const float* item_emb = (const float*)d_in[1];
    const float* Wg[3], *bg[3], *Wb[3], *bb[3];
    for (int k = 0; k < 3; ++k) {
        Wg[k] = (const float*)d_in[2 + 4 * k];
        bg[k] = (const float*)d_in[3 + 4 * k];
        Wb[k] = (const float*)d_in[4 + 4 * k];
        bb[k] = (const float*)d_in[5 + 4 * k];
    }
    const float* adj_vals = (const float*)d_in[14];
    const int*   adj_rows = (const int*)d_in[15];
    const int*   adj_cols = (const int*)d_in[16];
    float* out = (float*)d_out;

    float* ego  = (float*)d_ws;
    float* side = ego + (size_t)NTOT * DD;

    const size_t n4 = (size_t)NTOT * 16;                 // N*64 floats as float4
    const int TB = 256;
    const int initBlocks = (int)((n4 + TB - 1) / TB);
    const int zeroBlocks = initBlocks;
    const int spmmBlocks = (int)(((size_t)NEDGE * 32 + TB - 1) / TB);
    const int layerBlocks = (NTILES + 7) / 8;            // 8 tiles (waves) / block

    ngcf_init<<<initBlocks, TB, 0, stream>>>((const float4*)user_emb,
                                             (const float4*)item_emb,
                                             (float4*)ego, (float4*)out);
    for (int k = 0; k < 3; ++k) {
        ngcf_zero<<<zeroBlocks, TB, 0, stream>>>((float4*)side, n4);
        ngcf_spmm<<<spmmBlocks, TB, 0, stream>>>(adj_vals, adj_rows, adj_cols,
                                                 ego, side);
        ngcf_layer<<<layerBlocks, TB, 0, stream>>>(side, ego,
                                                   Wg[k], bg[k], Wb[k], bb[k],
                                                   out, k);
    }
}